// AE_32530082300068
// MI455X (gfx1250) — compile-verified
//
#include <hip/hip_runtime.h>
#include <hip/hip_bf16.h>

#define NUM_ITEMS 50000
#define NUM_USERS 16384
#define H0 512
#define LAT 256

typedef __attribute__((ext_vector_type(2))) float v2f;
typedef __attribute__((ext_vector_type(8))) float v8f;

// ---------------------------------------------------------------- init
__global__ void ae_zero_u32_kernel(unsigned int* __restrict__ p, int n) {
  int i = blockIdx.x * blockDim.x + threadIdx.x;
  if (i < n) p[i] = 0u;
}

// ---------------------------------------------------------------- stage 1a: user histogram
__global__ void ae_hist_kernel(const int* __restrict__ user,
                               int* __restrict__ counts, int n) {
  int i = blockIdx.x * blockDim.x + threadIdx.x;
  if (i < n) atomicAdd(&counts[user[i]], 1);
}

// ---------------------------------------------------------------- stage 1b: exclusive scan (single block)
// 16384 counts, 256 threads x 64 elements each; Hillis-Steele over partials in LDS.
__global__ void ae_scan_kernel(const int* __restrict__ counts,
                               int* __restrict__ offsets,
                               int* __restrict__ cursor) {
  __shared__ int partial[256];
  int t = threadIdx.x;
  int base = t * 64;
  int s = 0;
#pragma unroll 8
  for (int i = 0; i < 64; ++i) s += counts[base + i];
  partial[t] = s;
  __syncthreads();
  for (int off = 1; off < 256; off <<= 1) {
    int v = (t >= off) ? partial[t - off] : 0;
    __syncthreads();
    partial[t] += v;
    __syncthreads();
  }
  int run = (t == 0) ? 0 : partial[t - 1];  // exclusive offset of this chunk
  for (int i = 0; i < 64; ++i) {
    offsets[base + i] = run;
    cursor[base + i] = run;
    run += counts[base + i];
  }
  if (t == 255) offsets[NUM_USERS] = run;
}

// ---------------------------------------------------------------- stage 1c: scatter indices into user order
__global__ void ae_sort_kernel(const int* __restrict__ user,
                               int* __restrict__ cursor,
                               int* __restrict__ sorted, int n) {
  int i = blockIdx.x * blockDim.x + threadIdx.x;
  if (i < n) {
    int pos = atomicAdd(&cursor[user[i]], 1);
    sorted[pos] = i;
  }
}

// ---------------------------------------------------------------- stage 1d+2: per-user aggregation, fused tanh(agg + b_enc)
// one 256-thread block per user; thread owns columns h and h+256; no float atomics.
__global__ void ae_user_agg_kernel(const int* __restrict__ item,
                                   const float* __restrict__ rating,
                                   const int* __restrict__ sorted,
                                   const int* __restrict__ offsets,
                                   const float* __restrict__ Wenc,
                                   const float* __restrict__ benc,
                                   float* __restrict__ x) {
  int u = blockIdx.x;
  int t = threadIdx.x;
  int h0 = t, h1 = t + 256;
  const float* w0 = Wenc + (size_t)h0 * NUM_ITEMS;
  const float* w1 = Wenc + (size_t)h1 * NUM_ITEMS;
  float a0 = 0.0f, a1 = 0.0f;
  int beg = offsets[u], end = offsets[u + 1];
  for (int j = beg; j < end; ++j) {
    int i = sorted[j];
    int it = item[i];      // uniform across block -> broadcast from cache
    float r = rating[i];
    a0 += w0[it] * r;
    a1 += w1[it] * r;
  }
  x[(size_t)u * H0 + h0] = tanhf(a0 + benc[h0]);
  x[(size_t)u * H0 + h1] = tanhf(a1 + benc[h1]);
}

// ---------------------------------------------------------------- stages 3/4: C = tanh(A @ B^T + bias), fp32 WMMA
// 32x32 C tile per wave: 4 accumulators, 4 fragment loads feed 4 WMMAs per K-step
// (1 global_load_b64 per v_wmma). Grid exactly covers tiles -> EXEC all-1s.
__global__ void ae_gemm_tanh_wmma_kernel(const float* __restrict__ A, int lda,
                                         const float* __restrict__ B, int ldb,
                                         const float* __restrict__ bias,
                                         float* __restrict__ C, int ldc,
                                         int K, int ntiles_n) {
  int waveId = threadIdx.x >> 5;
  int lane = threadIdx.x & 31;
  int tile = blockIdx.x * (blockDim.x >> 5) + waveId;
  int m0 = (tile / ntiles_n) * 32;
  int n0 = (tile % ntiles_n) * 32;
  int row = lane & 15;   // A row (M) / B row (N) / C column within 16-tile
  int hi  = lane >> 4;   // K pair select: k + 2*hi, +1

  const float* a0p = A + (size_t)(m0 + row) * lda + 2 * hi;
  const float* a1p = a0p + (size_t)16 * lda;
  const float* b0p = B + (size_t)(n0 + row) * ldb + 2 * hi;
  const float* b1p = b0p + (size_t)16 * ldb;

  v8f c00 = {}, c01 = {}, c10 = {}, c11 = {};
  for (int k = 0; k < K; k += 4) {
    v2f a0 = *(const v2f*)(a0p + k);
    v2f a1 = *(const v2f*)(a1p + k);
    v2f b0 = *(const v2f*)(b0p + k);
    v2f b1 = *(const v2f*)(b1p + k);
    c00 = __builtin_amdgcn_wmma_f32_16x16x4_f32(false, a0, false, b0, (short)0, c00, false, false);
    c01 = __builtin_amdgcn_wmma_f32_16x16x4_f32(false, a0, false, b1, (short)0, c01, false, false);
    c10 = __builtin_amdgcn_wmma_f32_16x16x4_f32(false, a1, false, b0, (short)0, c10, false, false);
    c11 = __builtin_amdgcn_wmma_f32_16x16x4_f32(false, a1, false, b1, (short)0, c11, false, false);
  }

  float bv0 = bias[n0 + row];
  float bv1 = bias[n0 + 16 + row];
#pragma unroll
  for (int r = 0; r < 8; ++r) {
    int mA = m0 + 8 * hi + r;   // C layout: VGPR r -> M=r (lanes 0-15) / 8+r (lanes 16-31)
    int mB = mA + 16;
    C[(size_t)mA * ldc + (n0 + row)]      = tanhf(c00[r] + bv0);
    C[(size_t)mA * ldc + (n0 + 16 + row)] = tanhf(c01[r] + bv1);
    C[(size_t)mB * ldc + (n0 + row)]      = tanhf(c10[r] + bv0);
    C[(size_t)mB * ldc + (n0 + 16 + row)] = tanhf(c11[r] + bv1);
  }
}

// ---------------------------------------------------------------- stage 5: pred + loss
// one wave per target: 512-length dot of decoded[tu] and W_dec[ti]
__global__ void ae_decoder_dot_kernel(const int* __restrict__ tu,
                                      const int* __restrict__ ti,
                                      const float* __restrict__ tr,
                                      const float* __restrict__ decoded,
                                      const float* __restrict__ Wdec,
                                      const float* __restrict__ bdec,
                                      float* __restrict__ pred,
                                      float* __restrict__ loss_acc,
                                      int nt, float inv_nt) {
  __shared__ float lsum[8];
  int waveInBlock = threadIdx.x >> 5;
  int lane = threadIdx.x & 31;
  int t = blockIdx.x * (blockDim.x >> 5) + waveInBlock;

  float sq = 0.0f;
  if (t < nt) {
    const float4* g = (const float4*)(decoded + (size_t)tu[t] * H0) + lane;
    const float4* w = (const float4*)(Wdec + (size_t)ti[t] * H0) + lane;
    float acc = 0.0f;
#pragma unroll
    for (int j = 0; j < H0 / 128; ++j) {  // 4 x float4 per lane
      float4 gv = g[32 * j];
      float4 wv = w[32 * j];
      acc += gv.x * wv.x + gv.y * wv.y + gv.z * wv.z + gv.w * wv.w;
    }
#pragma unroll
    for (int off = 16; off; off >>= 1) acc += __shfl_xor(acc, off, 32);
    if (lane == 0) {
      float p = acc + bdec[ti[t]];
      pred[t] = p;
      float d = p - tr[t];
      sq = d * d * inv_nt;
    }
  }
  if (lane == 0) lsum[waveInBlock] = sq;
  __syncthreads();
  if (threadIdx.x == 0) {
    float s = 0.0f;
#pragma unroll
    for (int i = 0; i < 8; ++i) s += lsum[i];
    atomicAdd(loss_acc, s);  // one L2 atomic per 256-thread block
  }
}

// ---------------------------------------------------------------- launcher
extern "C" void kernel_launch(void* const* d_in, const int* in_sizes, int n_in,
                              void* d_out, int out_size, void* d_ws, size_t ws_size,
                              hipStream_t stream) {
  const int*   user   = (const int*)  d_in[0];
  const int*   item   = (const int*)  d_in[1];
  const float* rating = (const float*)d_in[2];
  const int*   t_user = (const int*)  d_in[3];
  const int*   t_item = (const int*)  d_in[4];
  const float* t_rate = (const float*)d_in[5];
  const float* W_enc  = (const float*)d_in[6];  // [H0, NUM_ITEMS]
  const float* b_enc  = (const float*)d_in[7];
  const float* W1     = (const float*)d_in[8];  // [LAT, H0]
  const float* b1     = (const float*)d_in[9];
  const float* W2     = (const float*)d_in[10]; // [H0, LAT]
  const float* b2     = (const float*)d_in[11];
  const float* W_dec  = (const float*)d_in[12]; // [NUM_ITEMS, H0]
  const float* b_dec  = (const float*)d_in[13];

  const int n  = in_sizes[0];
  const int nt = in_sizes[3];

  float* out = (float*)d_out;  // [nt] pred + [1] loss

  // workspace layout
  float* bufA    = (float*)d_ws;                           // 16384*512 f32: x -> decoded (32 MB)
  float* bufB    = bufA + (size_t)NUM_USERS * H0;          // 16384*256 f32: encoded (16 MB)
  int*   counts  = (int*)(bufB + (size_t)NUM_USERS * LAT); // [16384]
  int*   offsets = counts + NUM_USERS;                     // [16385]
  int*   cursor  = offsets + NUM_USERS + 1;                // [16384]
  int*   sorted  = cursor + NUM_USERS;                     // [n]

  // zero histogram + loss slot
  ae_zero_u32_kernel<<<(NUM_USERS + 255) / 256, 256, 0, stream>>>((unsigned int*)counts, NUM_USERS);
  ae_zero_u32_kernel<<<1, 32, 0, stream>>>((unsigned int*)(out + nt), 1);

  // counting sort by user (int atomics only)
  ae_hist_kernel<<<(n + 255) / 256, 256, 0, stream>>>(user, counts, n);
  ae_scan_kernel<<<1, 256, 0, stream>>>(counts, offsets, cursor);
  ae_sort_kernel<<<(n + 255) / 256, 256, 0, stream>>>(user, cursor, sorted, n);

  // per-user segment sum + fused tanh(.+b_enc) -> x in bufA (no float atomics)
  ae_user_agg_kernel<<<NUM_USERS, 256, 0, stream>>>(item, rating, sorted, offsets,
                                                    W_enc, b_enc, bufA);

  // encoded = tanh(x @ W1^T + b1)   M=16384, K=512, N=256 ; 32x32 tiles
  {
    int waves = (NUM_USERS / 32) * (LAT / 32);  // 512*8 = 4096
    ae_gemm_tanh_wmma_kernel<<<waves / 8, 256, 0, stream>>>(
        bufA, H0, W1, H0, b1, bufB, LAT, H0, LAT / 32);
  }

  // decoded = tanh(encoded @ W2^T + b2)   M=16384, K=256, N=512 ; 32x32 tiles
  {
    int waves = (NUM_USERS / 32) * (H0 / 32);   // 512*16 = 8192
    ae_gemm_tanh_wmma_kernel<<<waves / 8, 256, 0, stream>>>(
        bufB, LAT, W2, LAT, b2, bufA, H0, LAT, H0 / 32);
  }

  // pred + loss (wave per target)
  ae_decoder_dot_kernel<<<(nt + 7) / 8, 256, 0, stream>>>(
      t_user, t_item, t_rate, bufA, W_dec, b_dec, out, out + nt, nt, 1.0f / (float)nt);
}